// ReversePoseToWorldFrame_13494787244414
// MI455X (gfx1250) — compile-verified
//
#include <hip/hip_runtime.h>

// CDNA5 / gfx1250. wave32.
typedef __attribute__((ext_vector_type(2))) float v2f;
typedef __attribute__((ext_vector_type(8))) float v8f;

#define CS 8            // segments per chunk (K1 per-thread sequential product)
#define K1_THREADS 256  // chunks per K1 block

// ---------------- small 4x4 helpers (row-major, 16 floats) ----------------
__device__ __forceinline__ void mat_mul(const float* __restrict__ x,
                                        const float* __restrict__ y,
                                        float* __restrict__ r) {
#pragma unroll
  for (int i = 0; i < 4; ++i)
#pragma unroll
    for (int j = 0; j < 4; ++j)
      r[i * 4 + j] = x[i * 4 + 0] * y[0 * 4 + j] + x[i * 4 + 1] * y[1 * 4 + j] +
                     x[i * 4 + 2] * y[2 * 4 + j] + x[i * 4 + 3] * y[3 * 4 + j];
}

__device__ __forceinline__ void load16(const float* __restrict__ p, float* __restrict__ m) {
  const float4* q = (const float4*)p;
#pragma unroll
  for (int i = 0; i < 4; ++i) {
    float4 v = q[i];
    m[i * 4 + 0] = v.x; m[i * 4 + 1] = v.y; m[i * 4 + 2] = v.z; m[i * 4 + 3] = v.w;
  }
}

__device__ __forceinline__ void store16(float* __restrict__ p, const float* __restrict__ m) {
  float4* q = (float4*)p;
#pragma unroll
  for (int i = 0; i < 4; ++i)
    q[i] = make_float4(m[i * 4 + 0], m[i * 4 + 1], m[i * 4 + 2], m[i * 4 + 3]);
}

__device__ __forceinline__ void set_ident(float* m) {
#pragma unroll
  for (int i = 0; i < 16; ++i) m[i] = (i % 5 == 0) ? 1.0f : 0.0f;
}

// ---------------- K1: chunk products + block-local exclusive scan ----------------
// chunk c covers segments [c*CS, (c+1)*CS). boundary[s] = poses[s*8 + 7].
// Output: chunk_excl[c] = block-local exclusive product of chunks,
//         blocktot[blockIdx] = product of the block's 256 chunks.
__global__ void k1_chunk_scan(const float* __restrict__ poses,
                              float* __restrict__ chunk_excl,
                              float* __restrict__ blocktot) {
  // pad matrix stride to 17 floats: lane stride 17 is coprime with 64 banks
  __shared__ float sc[2][K1_THREADS * 17];
  const int t = threadIdx.x;
  const int c = blockIdx.x * K1_THREADS + t;

  float prod[16], tmp[16], nm[16];
  const int s0 = c * CS;
  load16(poses + (size_t)(s0 * 8 + 7) * 16, prod);
#pragma unroll
  for (int j = 1; j < CS; ++j) {
    load16(poses + (size_t)((s0 + j) * 8 + 7) * 16, nm);
    mat_mul(prod, nm, tmp);
#pragma unroll
    for (int i = 0; i < 16; ++i) prod[i] = tmp[i];
  }

  int buf = 0;
#pragma unroll
  for (int i = 0; i < 16; ++i) sc[0][t * 17 + i] = prod[i];

  // Hillis-Steele inclusive scan over 256 matrices (8 steps, left operand = earlier)
  for (int off = 1; off < K1_THREADS; off <<= 1) {
    __syncthreads();
    float cur[16], res[16];
#pragma unroll
    for (int i = 0; i < 16; ++i) cur[i] = sc[buf][t * 17 + i];
    if (t >= off) {
      float pre[16];
#pragma unroll
      for (int i = 0; i < 16; ++i) pre[i] = sc[buf][(t - off) * 17 + i];
      mat_mul(pre, cur, res);
    } else {
#pragma unroll
      for (int i = 0; i < 16; ++i) res[i] = cur[i];
    }
    __syncthreads();
#pragma unroll
    for (int i = 0; i < 16; ++i) sc[1 - buf][t * 17 + i] = res[i];
    buf ^= 1;
  }
  __syncthreads();

  float ex[16];
  if (t == 0) {
    set_ident(ex);
  } else {
#pragma unroll
    for (int i = 0; i < 16; ++i) ex[i] = sc[buf][(t - 1) * 17 + i];
  }
  store16(chunk_excl + (size_t)c * 16, ex);

  if (t == 0) {
    float tot[16];
#pragma unroll
    for (int i = 0; i < 16; ++i) tot[i] = sc[buf][(K1_THREADS - 1) * 17 + i];
    store16(blocktot + (size_t)blockIdx.x * 16, tot);
  }
}

// ---------------- K2: tiny sequential exclusive scan over block totals ----------------
__global__ void k2_block_scan(const float* __restrict__ blocktot,
                              float* __restrict__ blockexcl, int nb) {
  if (threadIdx.x != 0 || blockIdx.x != 0) return;
  float X[16], nm[16], tmp[16];
  set_ident(X);
  for (int b = 0; b < nb; ++b) {
    store16(blockexcl + (size_t)b * 16, X);
    load16(blocktot + (size_t)b * 16, nm);
    mat_mul(X, nm, tmp);
#pragma unroll
    for (int i = 0; i < 16; ++i) X[i] = tmp[i];
  }
}

// ---------------- K3: per-segment K = kf0 @ excl[s], LK = lf @ K ----------------
__global__ void k3_make_K(const float* __restrict__ poses,
                          const float* __restrict__ kfm,
                          const float* __restrict__ chunk_excl,
                          const float* __restrict__ blockexcl,
                          float* __restrict__ wsK, float* __restrict__ wsLK, int S) {
  const int s = blockIdx.x * blockDim.x + threadIdx.x;
  if (s >= S) return;
  const int c = s >> 3;        // CS == 8
  const int r = s & 7;
  const int blk = c >> 8;      // K1_THREADS == 256 chunks per K1 block

  float X[16], tmp[16], nm[16];
  load16(blockexcl + (size_t)blk * 16, X);
  load16(chunk_excl + (size_t)c * 16, nm);
  mat_mul(X, nm, tmp);
#pragma unroll
  for (int i = 0; i < 16; ++i) X[i] = tmp[i];

  for (int j = 0; j < r; ++j) {   // walk boundaries inside the chunk (L2-hot)
    load16(poses + (size_t)((c * 8 + j) * 8 + 7) * 16, nm);
    mat_mul(X, nm, tmp);
#pragma unroll
    for (int i = 0; i < 16; ++i) X[i] = tmp[i];
  }

  float kf[16];
  load16(kfm, kf);
  float K[16];
  mat_mul(kf, X, K);   // K[s] = kf0 @ exclusive_prefix

  // lf = [[R^T, -R^T t], [kf0 row 3]]
  float lf[16];
#pragma unroll
  for (int rr = 0; rr < 3; ++rr) {
#pragma unroll
    for (int cc = 0; cc < 3; ++cc) lf[rr * 4 + cc] = kf[cc * 4 + rr];
    lf[rr * 4 + 3] = -(kf[0 + rr] * kf[3] + kf[4 + rr] * kf[7] + kf[8 + rr] * kf[11]);
  }
#pragma unroll
  for (int cc = 0; cc < 4; ++cc) lf[12 + cc] = kf[12 + cc];

  float LK[16];
  mat_mul(lf, K, LK);  // LK @ E == lf @ (K @ E) == lf_poses

  store16(wsK + (size_t)s * 16, K);
  store16(wsLK + (size_t)s * 16, LK);
}

// ---------------- K4: streaming WMMA pass ----------------
// One wave handles a segment pair (a,b). One V_WMMA_F32_16X16X4_F32 per pose index p:
//   A(16x4) rows = [K_a; LK_a; K_b; LK_b],  B(4x16) cols = [E_a(p) | E_b(p) | don't-care]
// Useful D blocks:
//   D[ 0.. 3][0..3] = K_a @E_a = wf_a     -> lanes 0-3,   VGPRs 0-3
//   D[ 4.. 7][0..3] = LK_a@E_a = lfp_a    -> lanes 0-3,   VGPRs 4-7
//   D[ 8..11][4..7] = K_b @E_b = wf_b     -> lanes 20-23, VGPRs 0-3  (rows 8-15 live
//   D[12..15][4..7] = LK_b@E_b = lfp_b    -> lanes 20-23, VGPRs 4-7   on lanes 16-31)
// => every active store lane writes ALL 8 D registers with constant extract indices.
// A layout (ISA 7.12.2, 32-bit A 16x4): lane m<16 holds row m, K={0,1}; lane m+16 K={2,3}.
// B layout (row-striped): VGPR0 = rows k={0(l0-15),1(l16-31)}, VGPR1 = k={2,3}.
__global__ void k4_wmma_apply(const float* __restrict__ poses,
                              const float* __restrict__ wsK,
                              const float* __restrict__ wsLK,
                              float* __restrict__ out_wf,
                              float* __restrict__ out_lfp) {
  const int wave = (blockIdx.x * blockDim.x + threadIdx.x) >> 5;
  const int lane = threadIdx.x & 31;
  const int m = lane & 15;
  const int kh = lane >> 4;           // A K-half / B row-pair / D row-half
  const int a = wave * 2, b = wave * 2 + 1;

  // Prefetch the pair's 16 contiguous pose matrices (1KB): one line per lane.
  const char* pf = (const char*)(poses + (size_t)a * 8 * 16);
  __builtin_prefetch(pf + lane * 32, 0, 3);

  // ---- A operand: rows [K_a; LK_a; K_b; LK_b], per-lane float2 in WMMA layout ----
  const int sel = m >> 2, row = m & 3;
  const float* Ab;
  if (sel == 0)      Ab = wsK  + (size_t)a * 16;
  else if (sel == 1) Ab = wsLK + (size_t)a * 16;
  else if (sel == 2) Ab = wsK  + (size_t)b * 16;
  else               Ab = wsLK + (size_t)b * 16;
  const float2 a01 = *(const float2*)(Ab + row * 4 + 2 * kh);
  v2f A; A[0] = a01.x; A[1] = a01.y;

  // ---- B feed lanes: cols 0-3 <- E_a, cols 4-7 <- E_b, cols 8-15 unused ----
  const bool actB = (m < 8);
  const int  segB = (m < 4) ? a : b;
  const int  colB = m & 3;
  const float* EB = poses + (size_t)segB * 8 * 16;  // 8 poses of this segment

  // ---- store lanes: (m>>2)==kh  => lanes 0-3 (seg a) and 20-23 (seg b) ----
  const bool actS = ((m >> 2) == kh);
  const int  segS = kh ? b : a;
  const size_t outoff = (size_t)segS * 8 * 16 + (m & 3);

#pragma unroll
  for (int p = 0; p < 8; ++p) {
    float b0 = 0.0f, b1 = 0.0f;
    if (actB) {
      b0 = EB[p * 16 + (0 + kh) * 4 + colB];   // B VGPR0: k = kh
      b1 = EB[p * 16 + (2 + kh) * 4 + colB];   // B VGPR1: k = 2 + kh
    }
    v2f Bv; Bv[0] = b0; Bv[1] = b1;
    v8f C = {0.f, 0.f, 0.f, 0.f, 0.f, 0.f, 0.f, 0.f};
    // 8 args: (neg_a, A, neg_b, B, c_mod, C, reuse_a, reuse_b)
    v8f D = __builtin_amdgcn_wmma_f32_16x16x4_f32(false, A, false, Bv,
                                                  (short)0, C, false, false);
    if (actS) {
      float* dwf = out_wf  + outoff + p * 16;
      float* dlf = out_lfp + outoff + p * 16;
      __builtin_nontemporal_store(D[0], dwf + 0);   // wf rows 0..3 (this lane's col)
      __builtin_nontemporal_store(D[1], dwf + 4);
      __builtin_nontemporal_store(D[2], dwf + 8);
      __builtin_nontemporal_store(D[3], dwf + 12);
      __builtin_nontemporal_store(D[4], dlf + 0);   // lf_poses rows 0..3
      __builtin_nontemporal_store(D[5], dlf + 4);
      __builtin_nontemporal_store(D[6], dlf + 8);
      __builtin_nontemporal_store(D[7], dlf + 12);
    }
  }
}

// ---------------- host launch ----------------
extern "C" void kernel_launch(void* const* d_in, const int* in_sizes, int n_in,
                              void* d_out, int out_size, void* d_ws, size_t ws_size,
                              hipStream_t stream) {
  const float* poses = (const float*)d_in[0];       // (B,4,4) f32
  const float* kfm   = (const float*)d_in[1];       // (4,4) f32
  (void)n_in; (void)ws_size; (void)out_size;

  const int B  = in_sizes[0] / 16;                  // 1048576
  const int S  = B / 8;                             // 131072 segments (P == 8)
  const int NC = S / CS;                            // 16384 chunks
  const int NB = NC / K1_THREADS;                   // 64 K1 blocks

  float* w          = (float*)d_ws;
  float* chunk_excl = w;                            // NC * 16 floats   (1 MB)
  float* blocktot   = chunk_excl + (size_t)NC * 16; // NB * 16
  float* blockexcl  = blocktot + (size_t)NB * 16;   // NB * 16
  float* wsK        = blockexcl + (size_t)NB * 16;  // S * 16 floats    (8 MB)
  float* wsLK       = wsK + (size_t)S * 16;         // S * 16 floats    (8 MB)

  float* out_wf  = (float*)d_out;
  float* out_lfp = out_wf + (size_t)B * 16;

  k1_chunk_scan<<<NB, K1_THREADS, 0, stream>>>(poses, chunk_excl, blocktot);
  k2_block_scan<<<1, 64, 0, stream>>>(blocktot, blockexcl, NB);
  k3_make_K<<<(S + 255) / 256, 256, 0, stream>>>(poses, kfm, chunk_excl, blockexcl,
                                                 wsK, wsLK, S);
  const int pairs = S / 2;                          // 65536 waves, 8 waves/block
  k4_wmma_apply<<<pairs / 8, 256, 0, stream>>>(poses, wsK, wsLK, out_wf, out_lfp);
}